// LogDynamicRangeCompressionAdaptation_2422361555661
// MI455X (gfx1250) — compile-verified
//
#include <hip/hip_runtime.h>
#include <math.h>

typedef float v2f __attribute__((ext_vector_type(2)));
typedef float v8f __attribute__((ext_vector_type(8)));

#define T_LEN   160000
#define B_ROWS  128
#define TSUB    320            // chunk length (20 blocks of 16)
#define NCHUNK  500            // T_LEN / TSUB
#define NBLK    20             // TSUB / 16

// a = exp(-1/(16000*0.1)) = exp(-0.000625)
#define ALPHA_F   0.00062480472818370f      // 1 - a
#define L2A_F    (-9.0168440055560e-04f)    // log2(a)
#define A16_F     0.99004983374916805f      // a^16
#define ATSUB_F   0.81873075307798185f      // a^320
#define L2AT_F   (-0.28853900817779268f)    // log2(a^320)
#define INV_THR   1.0e5f
#define LN2_F     0.69314718055994531f

__device__ __forceinline__ float pow_a(float p) { return exp2f(p * L2A_F); }

// f = log(x_thr)/(x_thr-1), 0/0 -> 1.  Fast path: v_log_f32 + v_rcp_f32.
__device__ __forceinline__ float comp_f(float xv) {
    float xt = fmaxf(xv * INV_THR, 1.0f);
    float d  = xt - 1.0f;
    float f  = (__builtin_amdgcn_logf(xt) * LN2_F) * __builtin_amdgcn_rcpf(d);
    return (d == 0.0f) ? 1.0f : f;
}

// ---------------- Kernel 1: per-chunk decayed reduction -------------------
// S[row][chunk] = sum_k a^(TSUB-1-k) * (alpha * f(x[row][chunk*TSUB+k]))
__global__ void k1_chunk_reduce(const float* __restrict__ x, float* __restrict__ S) {
    int tid  = threadIdx.x;
    int lane = tid & 31;
    int wid  = blockIdx.x * 8 + (tid >> 5);       // one wave per (row, chunk)
    int row  = wid & (B_ROWS - 1);
    int chunk = wid >> 7;
    const float* xp = x + (size_t)row * T_LEN + (size_t)chunk * TSUB;
    float s = 0.0f;
#pragma unroll
    for (int i = 0; i < TSUB / 32; ++i) {
        int k = lane + 32 * i;
        float w = pow_a((float)(TSUB - 1 - k));
        s += w * comp_f(xp[k]);
    }
    s *= ALPHA_F;
#pragma unroll
    for (int off = 16; off > 0; off >>= 1) s += __shfl_xor(s, off, 32);
    if (lane == 0) S[(size_t)row * NCHUNK + chunk] = s;
}

// ---------------- Kernel 2: per-row scan over chunk carries ---------------
// carry_in[chunk][row]: state entering each chunk; zi = 1.
__global__ void k2_carry_scan(const float* __restrict__ S, float* __restrict__ carry) {
    __shared__ float sm[512];
    int row = blockIdx.x;
    int t   = threadIdx.x;
    float v = (t < NCHUNK) ? S[(size_t)row * NCHUNK + t] : 0.0f;
    sm[t] = v;
    __syncthreads();
    float wd = ATSUB_F;                           // a^(TSUB * 2^step)
#pragma unroll
    for (int d = 1; d < 512; d <<= 1) {
        float add = (t >= d) ? sm[t - d] : 0.0f;
        __syncthreads();
        v += wd * add;
        sm[t] = v;
        __syncthreads();
        wd *= wd;
    }
    if (t < NCHUNK) {
        float prev = (t == 0) ? 0.0f : sm[t - 1];
        float ci = exp2f((float)t * L2AT_F) + prev;   // zi=1 decayed + accumulated
        carry[(size_t)t * B_ROWS + row] = ci;
    }
}

// ---------------- Kernel 3: WMMA intra-chunk scan + apply -----------------
// For a 16-row x 16-time block: local[m][t] = sum_j b[m][j] * a^(t-j) (j<=t)
//   => D = A(data 16xK) x B(L^T Kx16) accumulated over 4 K=4 WMMA steps.
// g[m][t] = local[m][t] + a^(t+1) * carry[m];  out = x * g * 1e5.
__global__ void k3_scan_apply(const float* __restrict__ x,
                              const float* __restrict__ carry,
                              float* __restrict__ out) {
    int tid  = threadIdx.x;
    int lane = tid & 31;
    int half = lane >> 4;                         // 0: lanes 0-15, 1: lanes 16-31
    int col  = lane & 15;                         // N index (time) / A row (batch)
    int wid  = blockIdx.x * 8 + (tid >> 5);
    int rowtile = wid & 7;
    int chunk   = wid >> 3;
    int row0 = rowtile * 16;

    // Constant B operand: B[j][t] = a^(t-j) for t>=j else 0.
    // 32-bit B 4x16 layout: reg.x -> K = 2*half, reg.y -> K = 2*half+1, col = lane&15.
    v2f bmat[4];
#pragma unroll
    for (int p = 0; p < 4; ++p) {
        int j0 = 4 * p + 2 * half;
        bmat[p].x = (col >= j0)     ? pow_a((float)(col - j0))     : 0.0f;
        bmat[p].y = (col >= j0 + 1) ? pow_a((float)(col - j0 - 1)) : 0.0f;
    }
    float pa_out = pow_a((float)(col + 1));       // a^(t+1) for this lane's output col

    // Carry state entering this chunk, per D row owned by this lane half.
    float c[8];
#pragma unroll
    for (int r = 0; r < 8; ++r)
        c[r] = carry[(size_t)chunk * B_ROWS + row0 + half * 8 + r];

    // A-layout pointer: lane holds batch row m = (lane&15); K offset 2*half.
    const float* xA = x + (size_t)(row0 + col) * T_LEN + (size_t)chunk * TSUB;
    // D-layout pointers: VGPR r -> row row0 + half*8 + r, column = col.
    const size_t dbase = (size_t)(row0 + half * 8) * T_LEN + (size_t)chunk * TSUB + col;
    const float* xD = x + dbase;
    float*       oD = out + dbase;

    for (int blk = 0; blk < NBLK; ++blk) {
        int t0 = blk * 16;
        v8f acc = {};
#pragma unroll
        for (int p = 0; p < 4; ++p) {
            v2f xv = *reinterpret_cast<const v2f*>(xA + t0 + 4 * p + 2 * half);
            v2f av;
            av.x = ALPHA_F * comp_f(xv.x);
            av.y = ALPHA_F * comp_f(xv.y);
            acc = __builtin_amdgcn_wmma_f32_16x16x4_f32(false, av, false, bmat[p],
                                                        (short)0, acc, false, false);
        }
#pragma unroll
        for (int r = 0; r < 8; ++r) {
            float local = acc[r];
            float g = local + pa_out * c[r];
            float xv = xD[(size_t)r * T_LEN + t0];
            oD[(size_t)r * T_LEN + t0] = xv * g * INV_THR;
            // carry update: column 15 of this row, broadcast within each half-wave
            float last = __shfl(local, 15, 16);
            c[r] = last + A16_F * c[r];
        }
    }
}

extern "C" void kernel_launch(void* const* d_in, const int* in_sizes, int n_in,
                              void* d_out, int out_size, void* d_ws, size_t ws_size,
                              hipStream_t stream) {
    (void)in_sizes; (void)n_in; (void)out_size; (void)ws_size;
    const float* x = (const float*)d_in[0];
    float* out = (float*)d_out;
    float* S     = (float*)d_ws;                       // [B_ROWS][NCHUNK]
    float* carry = S + (size_t)B_ROWS * NCHUNK;        // [NCHUNK][B_ROWS]

    // 1) per-chunk reductions: one wave per (row, chunk)
    k1_chunk_reduce<<<(B_ROWS * NCHUNK) / 8, 256, 0, stream>>>(x, S);
    // 2) per-row parallel scan over 500 chunk carries
    k2_carry_scan<<<B_ROWS, 512, 0, stream>>>(S, carry);
    // 3) WMMA-based intra-chunk scan + output: one wave per (rowtile, chunk)
    k3_scan_apply<<<(8 * NCHUNK) / 8, 256, 0, stream>>>(x, carry, out);
}